// get_CostVolume_75024488727085
// MI455X (gfx1250) — compile-verified
//
#include <hip/hip_runtime.h>
#include <cstdint>

#define D_DIM 9
#define A_DIM 9
#define C_DIM 16
#define H_DIM 256
#define W_DIM 256

// Native clang vector (required by __builtin_nontemporal_store; HIP's float4 is a class).
typedef __attribute__((ext_vector_type(4))) float vf4;

// CDNA5 async copy: global memory -> LDS, 16 bytes per lane, tracked by ASYNCcnt.
__device__ __forceinline__ void async_load_b128(const float* gsrc, const float* ldst) {
    // Low 32 bits of a generic pointer into the shared aperture are the LDS byte address.
    unsigned lds_addr = (unsigned)(uintptr_t)ldst;
    asm volatile("global_load_async_to_lds_b128 %0, %1, off"
                 :: "v"(lds_addr), "v"(gsrc)
                 : "memory");
}

__device__ __forceinline__ void wait_async_all() {
    asm volatile("s_wait_asynccnt 0" ::: "memory");
}

// Non-temporal 128-bit store: output is write-once streamed data (340 MB);
// keep it from evicting the 36 MB of 9x-reused input rows out of the 192 MB L2.
__device__ __forceinline__ void store_nt_f4(float* dst, vf4 v) {
    __builtin_nontemporal_store(v, (vf4*)dst);
}

__global__ __launch_bounds__(256) void cost_volume_kernel(
        const float* __restrict__ in, float* __restrict__ out) {
    const int y   = blockIdx.x;   // 0..255
    const int a   = blockIdx.y;   // 0..8
    const int d   = blockIdx.z;   // 0..8
    const int tid = threadIdx.x;  // 0..255

    const int t  = (d - D_DIM / 2) * (a - A_DIM / 2);  // |t| <= 16
    const int yy = y - t;

    const int lane_c = tid >> 6;          // 0..3 : channel sub-index
    const int x4     = (tid & 63) << 2;   // 0,4,...,252 : float4 column

    // out[0, d, c, a, y, x] : flat = ((((d*16)+c)*9 + a)*256 + y)*256 + x
    const size_t cStrideOut = (size_t)A_DIM * H_DIM * W_DIM;  // 589824
    float* outBase = out + (((size_t)d * C_DIM) * A_DIM + a) * ((size_t)H_DIM * W_DIM)
                         + (size_t)y * W_DIM;

    if (yy < 0 || yy >= H_DIM) {
        // Whole (d,a,y) slab is masked out: 16 channels x 256 floats of zeros.
        const vf4 z = {0.f, 0.f, 0.f, 0.f};
        #pragma unroll
        for (int cg = 0; cg < 4; ++cg) {
            const int c = (cg << 2) + lane_c;
            store_nt_f4(outBase + (size_t)c * cStrideOut + x4, z);
        }
        return;
    }

    // Stage in[c, a, yy, 0..255] for all 16 channels into LDS (16 KB) via async DMA.
    __shared__ __align__(16) float smem[C_DIM * W_DIM];

    const size_t cStrideIn = (size_t)A_DIM * H_DIM * W_DIM;           // 589824
    const float* gRow = in + ((size_t)a * H_DIM + yy) * W_DIM;        // c = 0 row

    #pragma unroll
    for (int k = 0; k < 4; ++k) {
        const int f  = tid + (k << 8);     // float4 slot 0..1023
        const int c  = f >> 6;             // 0..15
        const int xo = (f & 63) << 2;      // 0..252
        async_load_b128(gRow + (size_t)c * cStrideIn + xo, &smem[c * W_DIM + xo]);
    }
    wait_async_all();
    __syncthreads();

    // Emit shifted, x-masked rows: aligned b128 global stores, LDS absorbs misalignment.
    #pragma unroll
    for (int cg = 0; cg < 4; ++cg) {
        const int c = (cg << 2) + lane_c;
        const float* s = &smem[c * W_DIM];
        vf4 v;
        #pragma unroll
        for (int j = 0; j < 4; ++j) {
            const int xx = x4 + j - t;
            const int xc = xx < 0 ? 0 : (xx > (W_DIM - 1) ? (W_DIM - 1) : xx);
            const float m = (xx == xc) ? 1.0f : 0.0f;
            v[j] = s[xc] * m;
        }
        store_nt_f4(outBase + (size_t)c * cStrideOut + x4, v);
    }
}

extern "C" void kernel_launch(void* const* d_in, const int* in_sizes, int n_in,
                              void* d_out, int out_size, void* d_ws, size_t ws_size,
                              hipStream_t stream) {
    const float* in = (const float*)d_in[0];
    float* out = (float*)d_out;
    dim3 grid(H_DIM, A_DIM, D_DIM);   // (y, a, d) = 256 x 9 x 9 blocks
    cost_volume_kernel<<<grid, dim3(256), 0, stream>>>(in, out);
}